// DDNLoss_57939108823299
// MI455X (gfx1250) — compile-verified
//
#include <hip/hip_runtime.h>
#include <hip/hip_bf16.h>

typedef __attribute__((ext_vector_type(2))) float v2f;
typedef __attribute__((ext_vector_type(8))) float v8f;

#define ALPHA_F 0.25f
#define FG_WEIGHT 13.0f
#define BG_WEIGHT 1.0f
#define C_DIM 81
#define H_DIM 112
#define W_DIM 200
#define BN_DIM 12
#define M_BOX 20
#define HW_DIM (H_DIM * W_DIM)                  // 22400
#define W4_DIM (W_DIM / 4)                      // 50
#define GROUPS (BN_DIM * H_DIM * W4_DIM)        // 67200
#define NPIX ((float)(BN_DIM * HW_DIM))         // 268800
#define BLOCK_SZ 256

// Wave-level sum of (a0 + a1) across all 32 lanes using the f32 WMMA unit.
// A (16x4 f32) = per-lane float2 accumulators; B = ones => D rows = row-sums.
// d[0..7] at any lane in [0,15] sum rows 0..7; lanes [16,31] sum rows 8..15.
__device__ __forceinline__ float wave_sum_wmma(float a0, float a1) {
    v2f a; a.x = a0; a.y = a1;
    v2f b; b.x = 1.0f; b.y = 1.0f;
    v8f c = {};
    v8f d = __builtin_amdgcn_wmma_f32_16x16x4_f32(
        /*neg_a=*/false, a, /*neg_b=*/false, b,
        /*c_mod=*/(short)0, c, /*reuse_a=*/false, /*reuse_b=*/false);
    float p = d[0] + d[1] + d[2] + d[3] + d[4] + d[5] + d[6] + d[7];
    p += __shfl_xor(p, 16, 32);   // rows 0..7 half + rows 8..15 half
    return p;
}

__global__ __launch_bounds__(BLOCK_SZ)
void ddn_loss_main(const float* __restrict__ logits,
                   const int*   __restrict__ target,
                   const float* __restrict__ boxes,
                   float*       __restrict__ partial) {
    const int g = blockIdx.x * BLOCK_SZ + threadIdx.x;

    float acc0 = 0.0f, acc1 = 0.0f;

    if (g < GROUPS) {
        const int w4  = (g % W4_DIM) * 4;
        const int row = g / W4_DIM;      // bn*H + h
        const int h   = row % H_DIM;
        const int bn  = row / H_DIM;

        const size_t pix_off = (size_t)bn * HW_DIM + (size_t)h * W_DIM + w4;
        const float* base = logits + (size_t)bn * C_DIM * HW_DIM
                                   + (size_t)h * W_DIM + w4;

        const int4 t4 = *(const int4*)(target + pix_off);
        const int tt[4] = { t4.x, t4.y, t4.z, t4.w };

        // Online softmax state for 4 pixels
        float m[4]  = { -3.4028235e38f, -3.4028235e38f, -3.4028235e38f, -3.4028235e38f };
        float s[4]  = { 0.0f, 0.0f, 0.0f, 0.0f };
        float xt[4] = { 0.0f, 0.0f, 0.0f, 0.0f };

        #pragma unroll 3
        for (int c = 0; c < C_DIM; ++c) {
            const float4 v4 = *(const float4*)(base + (size_t)c * HW_DIM);
            __builtin_prefetch(base + (size_t)(c + 8) * HW_DIM, 0, 0);
            const float vv[4] = { v4.x, v4.y, v4.z, v4.w };
            #pragma unroll
            for (int j = 0; j < 4; ++j) {
                const float v  = vv[j];
                const float nm = fmaxf(m[j], v);
                s[j] = s[j] * __expf(m[j] - nm) + __expf(v - nm);
                m[j] = nm;
                xt[j] = (c == tt[j]) ? v : xt[j];
            }
        }

        // Foreground mask from 20 boxes (x, y, w, h) at 1/8 resolution
        bool fg[4] = { false, false, false, false };
        const float hf = (float)h;
        const float4* bx = (const float4*)(boxes + (size_t)bn * M_BOX * 4);
        #pragma unroll 4
        for (int mi = 0; mi < M_BOX; ++mi) {
            const float4 bb = bx[mi];
            const float u1 = floorf(bb.x * 0.125f);
            const float v1 = floorf(bb.y * 0.125f);
            const float u2 = ceilf((bb.x + bb.z) * 0.125f);
            const float v2 = ceilf((bb.y + bb.w) * 0.125f);
            const bool iny = (hf >= v1) && (hf < v2);
            #pragma unroll
            for (int j = 0; j < 4; ++j) {
                const float wf = (float)(w4 + j);
                fg[j] = fg[j] || (iny && (wf >= u1) && (wf < u2));
            }
        }

        float wl[4];
        #pragma unroll
        for (int j = 0; j < 4; ++j) {
            const float log_pt = xt[j] - m[j] - __logf(s[j]);
            const float pt     = __expf(log_pt);
            const float om     = 1.0f - pt;
            const float loss   = -ALPHA_F * om * om * log_pt;
            wl[j] = loss * (fg[j] ? FG_WEIGHT : BG_WEIGHT);
        }
        acc0 = wl[0] + wl[1];
        acc1 = wl[2] + wl[3];
    }

    // Wave reduction via WMMA (EXEC is all-ones here: no early returns)
    const float wsum = wave_sum_wmma(acc0, acc1);

    __shared__ float lds[BLOCK_SZ / 32];
    const int wid  = threadIdx.x >> 5;
    const int lane = threadIdx.x & 31;
    if (lane == 0) lds[wid] = wsum;
    __syncthreads();
    if (threadIdx.x == 0) {
        float t = 0.0f;
        #pragma unroll
        for (int i = 0; i < BLOCK_SZ / 32; ++i) t += lds[i];
        partial[blockIdx.x] = t;
    }
}

__global__ __launch_bounds__(BLOCK_SZ)
void ddn_loss_finish(const float* __restrict__ partial, int n,
                     float* __restrict__ out) {
    float acc = 0.0f;
    for (int i = threadIdx.x; i < n; i += BLOCK_SZ) acc += partial[i];

    const float wsum = wave_sum_wmma(acc, 0.0f);

    __shared__ float lds[BLOCK_SZ / 32];
    const int wid  = threadIdx.x >> 5;
    const int lane = threadIdx.x & 31;
    if (lane == 0) lds[wid] = wsum;
    __syncthreads();
    if (threadIdx.x == 0) {
        float t = 0.0f;
        #pragma unroll
        for (int i = 0; i < BLOCK_SZ / 32; ++i) t += lds[i];
        out[0] = t * (1.0f / NPIX);
    }
}

extern "C" void kernel_launch(void* const* d_in, const int* in_sizes, int n_in,
                              void* d_out, int out_size, void* d_ws, size_t ws_size,
                              hipStream_t stream) {
    const float* logits = (const float*)d_in[0];   // [2,6,81,112,200] f32
    const int*   target = (const int*)d_in[1];     // [2,6,112,200] i32
    const float* boxes  = (const float*)d_in[2];   // [2,6,20,4] f32
    float* out = (float*)d_out;
    float* partial = (float*)d_ws;

    const int nblocks = (GROUPS + BLOCK_SZ - 1) / BLOCK_SZ;   // 263
    ddn_loss_main<<<nblocks, BLOCK_SZ, 0, stream>>>(logits, target, boxes, partial);
    ddn_loss_finish<<<1, BLOCK_SZ, 0, stream>>>(partial, nblocks, out);
}